// GraphConvolutionK_25752623907270
// MI455X (gfx1250) — compile-verified
//
#include <hip/hip_runtime.h>
#include <hip/hip_bf16.h>

typedef __attribute__((ext_vector_type(2))) float v2f;
typedef __attribute__((ext_vector_type(8))) float v8f;

#define D_IN   128
#define D_OUT  32

// ---------------------------------------------------------------------------
// Stage 1: H[M x 32] = X[M x 128] @ W[128 x 32] using V_WMMA_F32_16X16X4_F32.
// One wave32 computes a 16x32 output tile (two 16x16 accumulators).
// A-frag layout (32-bit A 16x4): lane%16 = M-row; lanes 0-15 hold K=kk,kk+1
// in VGPR0/1, lanes 16-31 hold K=kk+2,kk+3 -> a contiguous float2 load.
// B-frag: lane%16 = N-col; same K striping by lane half.
// C/D layout: VGPR i -> M = i (+8 for lanes 16-31), N = lane%16.
// ---------------------------------------------------------------------------
__global__ void __launch_bounds__(256)
gemm_xw_wmma(const float* __restrict__ X, const float* __restrict__ W,
             float* __restrict__ H, int Mrows) {
  const int lane = threadIdx.x & 31;
  const int wave = threadIdx.x >> 5;
  const int tile = blockIdx.x * (blockDim.x >> 5) + wave;
  if (tile * 16 >= Mrows) return;              // wave-uniform: EXEC stays all-1s

  const int m0   = tile * 16;
  const int row  = m0 + (lane & 15);           // A: M-row for this lane
  const int koff = (lane >> 4) << 1;           // 0 for lanes 0-15, 2 for 16-31
  const int ncol = lane & 15;                  // B/C/D: N-col for this lane

  v8f acc0 = {0.f, 0.f, 0.f, 0.f, 0.f, 0.f, 0.f, 0.f};   // N = 0..15
  v8f acc1 = {0.f, 0.f, 0.f, 0.f, 0.f, 0.f, 0.f, 0.f};   // N = 16..31

  const float* xrow = X + (size_t)row * D_IN;

#pragma unroll 4
  for (int kk = 0; kk < D_IN; kk += 4) {
    // A fragment: X[row][kk+koff], X[row][kk+koff+1]  (contiguous)
    v2f a = *(const v2f*)(xrow + kk + koff);

    // B fragments: W[(kk+koff)..+1][ncol] and [16+ncol]
    const float* wr0 = W + (size_t)(kk + koff) * D_OUT;
    v2f b0, b1;
    b0.x = wr0[ncol];
    b0.y = wr0[D_OUT + ncol];
    b1.x = wr0[16 + ncol];
    b1.y = wr0[D_OUT + 16 + ncol];

    acc0 = __builtin_amdgcn_wmma_f32_16x16x4_f32(false, a, false, b0,
                                                 (short)0, acc0, false, false);
    acc1 = __builtin_amdgcn_wmma_f32_16x16x4_f32(false, a, false, b1,
                                                 (short)0, acc1, false, false);
  }

  // Store D: VGPR i -> row m0 + i (+8 for upper lane half), col = ncol (+16)
  const int crow = m0 + ((lane >> 4) << 3);
  float* hrow = H + (size_t)crow * D_OUT;
#pragma unroll
  for (int i = 0; i < 8; ++i) {
    hrow[(size_t)i * D_OUT + ncol]      = acc0[i];
    hrow[(size_t)i * D_OUT + 16 + ncol] = acc1[i];
  }
}

// ---------------------------------------------------------------------------
// Stage 2: edge scatter. One block per edge; each of 256 threads moves a
// float4 of the 1024-float (K*D_OUT) message via native f32 atomics.
// h lives in d_ws (41 MB) and stays L2-resident (192 MB L2).
// ---------------------------------------------------------------------------
__global__ void __launch_bounds__(256)
scatter_edges(const float* __restrict__ H, const int* __restrict__ erow,
              const int* __restrict__ ecol, const float* __restrict__ eval,
              float* __restrict__ out, int E) {
  const int e = blockIdx.x;
  if (e >= E) return;
  const int   r = erow[e];
  const int   c = ecol[e];
  const float v = eval[e];

  const float4* src = (const float4*)(H + (size_t)c * (32 * D_OUT));
  float*        dst = out + (size_t)r * (32 * D_OUT);

  const int t = threadIdx.x;           // 0..255 -> 256 * 4 = 1024 floats
  float4 d = src[t];
  const int base = t * 4;
  unsafeAtomicAdd(dst + base + 0, d.x * v);
  unsafeAtomicAdd(dst + base + 1, d.y * v);
  unsafeAtomicAdd(dst + base + 2, d.z * v);
  unsafeAtomicAdd(dst + base + 3, d.w * v);
}

// ---------------------------------------------------------------------------
// Stage 3: in-place ReLU, float4-vectorized (out_size divisible by 4).
// ---------------------------------------------------------------------------
__global__ void __launch_bounds__(256)
relu_kernel(float4* __restrict__ out, int n4) {
  int i = blockIdx.x * blockDim.x + threadIdx.x;
  if (i >= n4) return;
  float4 d = out[i];
  d.x = d.x > 0.f ? d.x : 0.f;
  d.y = d.y > 0.f ? d.y : 0.f;
  d.z = d.z > 0.f ? d.z : 0.f;
  d.w = d.w > 0.f ? d.w : 0.f;
  out[i] = d;
}

extern "C" void kernel_launch(void* const* d_in, const int* in_sizes, int n_in,
                              void* d_out, int out_size, void* d_ws, size_t ws_size,
                              hipStream_t stream) {
  const float* x    = (const float*)d_in[0];   // [N, K, 128] f32
  const float* w    = (const float*)d_in[1];   // [128, 32]   f32
  const int*   erow = (const int*)  d_in[2];   // [E]
  const int*   ecol = (const int*)  d_in[3];   // [E]
  const float* eval = (const float*)d_in[4];   // [E]
  float*       out  = (float*)d_out;           // [N, K, 32]  f32
  float*       h    = (float*)d_ws;            // [N*K, 32] scratch (~41 MB)

  const int M = in_sizes[0] / D_IN;            // N*K = 320000 rows
  const int E = in_sizes[2];                   // 160000 edges

  // 1) H = X @ W   (WMMA f32; 16-row tiles, 8 waves/block)
  const int tiles  = M / 16;                   // 320000 % 16 == 0
  const int blocks = (tiles + 7) / 8;
  gemm_xw_wmma<<<blocks, 256, 0, stream>>>(x, w, h, M);

  // 2) zero the poisoned output before atomic accumulation
  hipMemsetAsync(d_out, 0, (size_t)out_size * sizeof(float), stream);

  // 3) out[row] += val * H[col]  via native f32 atomics
  scatter_edges<<<E, 256, 0, stream>>>(h, erow, ecol, eval, out, E);

  // 4) ReLU in place
  const int n4 = out_size / 4;
  relu_kernel<<<(n4 + 255) / 256, 256, 0, stream>>>((float4*)d_out, n4);
}